// LocalTopologicalStructureContrastive_49649821942406
// MI455X (gfx1250) — compile-verified
//
#include <hip/hip_runtime.h>
#include <hip/hip_bf16.h>

typedef __attribute__((ext_vector_type(2))) float v2f;
typedef __attribute__((ext_vector_type(8))) float v8f;

#define D_DIM 128
#define INV_TAU 10.0f
#define JCHUNKS 8
#define LDS_STRIDE 132   // 128 + 4 pad floats -> 4-bank advance per row, conflict-free
#define TILE_ROWS 16

// ---------------- init: zero accumulators + output ----------------
__global__ void lts_init_kernel(float* pos_sum, float* pos_cnt, float* out, int N) {
    int i = blockIdx.x * blockDim.x + threadIdx.x;
    if (i < N) { pos_sum[i] = 0.0f; pos_cnt[i] = 0.0f; }
    if (i == 0) out[0] = 0.0f;
}

// ---------------- normalize: one wave32 per row ----------------
__global__ __launch_bounds__(256)
void lts_normalize_kernel(const float* __restrict__ Z, float* __restrict__ Zn,
                          float* __restrict__ diag, int N) {
    const int lane = threadIdx.x & 31;
    const int wave = threadIdx.x >> 5;
    const int row  = blockIdx.x * 8 + wave;
    if (row >= N) return;
    const float4 z = ((const float4*)(Z + (size_t)row * D_DIM))[lane];
    float ss = z.x * z.x + z.y * z.y + z.z * z.z + z.w * z.w;
    ss += __shfl_xor(ss, 16);
    ss += __shfl_xor(ss, 8);
    ss += __shfl_xor(ss, 4);
    ss += __shfl_xor(ss, 2);
    ss += __shfl_xor(ss, 1);
    const float inv = rsqrtf(ss);
    float4 zn; zn.x = z.x * inv; zn.y = z.y * inv; zn.z = z.z * inv; zn.w = z.w * inv;
    ((float4*)(Zn + (size_t)row * D_DIM))[lane] = zn;
    if (lane == 0) diag[row] = __expf(INV_TAU * (ss * inv * inv));
}

// ---------------- positive pairs: one wave32 per pair ----------------
__global__ __launch_bounds__(256)
void lts_pairs_kernel(const float* __restrict__ Zn, const int* __restrict__ pairs,
                      float* pos_sum, float* pos_cnt, int N, int P) {
    const int lane = threadIdx.x & 31;
    const int p    = (blockIdx.x * blockDim.x + threadIdx.x) >> 5;
    if (p >= P) return;
    const int a = pairs[2 * p];
    const int b = pairs[2 * p + 1];
    const bool valid = (b >= 0) && (b < N) && (a >= 0) && (a < N);
    float d = 0.0f;
    if (valid) {
        const float4 za = ((const float4*)(Zn + (size_t)a * D_DIM))[lane];
        const float4 zb = ((const float4*)(Zn + (size_t)b * D_DIM))[lane];
        d = za.x * zb.x + za.y * zb.y + za.z * zb.z + za.w * zb.w;
    }
    d += __shfl_xor(d, 16);
    d += __shfl_xor(d, 8);
    d += __shfl_xor(d, 4);
    d += __shfl_xor(d, 2);
    d += __shfl_xor(d, 1);
    if (lane == 0 && valid) {
        atomicAdd(&pos_sum[a], __expf(d * INV_TAU));
        atomicAdd(&pos_cnt[a], 1.0f);
    }
}

// ---------------- Gram row-sums via V_WMMA_F32_16X16X4_F32 ----------------
// grid = (N/128 row strips, JCHUNKS column chunks); block = 256 = 8 waves.
// B tile (16 cols x 128 K = 8KB) is shared by all 8 waves -> staged in LDS,
// double-buffered: issue next tile's coalesced b128 global loads, compute the
// 32-WMMA chain on the current buffer from ds_load_b64 (padded stride, zero
// bank conflicts), then ds_store_b128 the next tile and barrier once/tile.
__global__ __launch_bounds__(256)
void lts_gram_rowsum_kernel(const float* __restrict__ Zn,
                            float* __restrict__ rowsum_part, int N) {
    __shared__ float Btile[2][TILE_ROWS * LDS_STRIDE];

    const int tid  = threadIdx.x;
    const int lane = tid & 31;
    const int wave = tid >> 5;
    const int m0   = blockIdx.x * 128 + wave * 16;
    const int mrow = lane & 15;        // matrix row within 16-row fragment
    const int koff = (lane >> 4) * 2;  // K-pair offset per ISA 16x4 f32 layout
    const int jspan  = N / JCHUNKS;
    const int jbeg   = blockIdx.y * jspan;
    const int ntiles = jspan / TILE_ROWS;

    // staging indices: each thread moves two float4 chunks of the 16x128 tile
    const int r0 = tid >> 5,          c0 = (tid & 31) * 4;
    const int r1 = (tid + 256) >> 5,  c1 = c0;

    // A fragments: 16 rows x 128 K preloaded into 32 v2f registers
    const float* arow = Zn + (size_t)(m0 + mrow) * D_DIM + koff;
    v2f a[32];
#pragma unroll
    for (int kk = 0; kk < 32; ++kk)
        a[kk] = *(const v2f*)(arow + kk * 4);

    float rsum[8];
#pragma unroll
    for (int r = 0; r < 8; ++r) rsum[r] = 0.0f;

    // prologue: stage tile 0 into buffer 0
    {
        const float* src = Zn + (size_t)jbeg * D_DIM;
        float4 t0 = *(const float4*)(src + r0 * D_DIM + c0);
        float4 t1 = *(const float4*)(src + r1 * D_DIM + c1);
        *(float4*)(&Btile[0][r0 * LDS_STRIDE + c0]) = t0;
        *(float4*)(&Btile[0][r1 * LDS_STRIDE + c1]) = t1;
    }
    __syncthreads();

    for (int jt = 0; jt < ntiles; ++jt) {
        const int  cur  = jt & 1;
        const bool more = (jt + 1) < ntiles;   // uniform branch: EXEC stays full

        // issue next tile's global loads early (latency hidden by WMMA chain)
        float4 t0 = {}, t1 = {};
        if (more) {
            const float* src = Zn + (size_t)(jbeg + (jt + 1) * TILE_ROWS) * D_DIM;
            t0 = *(const float4*)(src + r0 * D_DIM + c0);
            t1 = *(const float4*)(src + r1 * D_DIM + c1);
        }

        // compute on current LDS buffer
        const float* bbase = &Btile[cur][mrow * LDS_STRIDE + koff];
        v8f c = {};
#pragma unroll
        for (int kk = 0; kk < 32; ++kk) {
            v2f b = *(const v2f*)(bbase + kk * 4);   // ds_load_b64, conflict-free
            c = __builtin_amdgcn_wmma_f32_16x16x4_f32(false, a[kk], false, b,
                                                      (short)0, c, false, false);
        }
#pragma unroll
        for (int r = 0; r < 8; ++r) rsum[r] += __expf(c[r] * INV_TAU);

        if (more) {
            *(float4*)(&Btile[cur ^ 1][r0 * LDS_STRIDE + c0]) = t0;
            *(float4*)(&Btile[cur ^ 1][r1 * LDS_STRIDE + c1]) = t1;
        }
        __syncthreads();
    }

    // C/D layout: VGPR r, lane l -> element (M = r + 8*(l>>4), Ncol = l&15).
#pragma unroll
    for (int r = 0; r < 8; ++r) {
        float v = rsum[r];
        v += __shfl_xor(v, 1);
        v += __shfl_xor(v, 2);
        v += __shfl_xor(v, 4);
        v += __shfl_xor(v, 8);
        if ((lane & 15) == 0)
            rowsum_part[(size_t)blockIdx.y * N + m0 + (lane >> 4) * 8 + r] = v;
    }
}

// ---------------- finalize: per-row loss, reduce to scalar ----------------
__global__ __launch_bounds__(256)
void lts_finalize_kernel(const float* __restrict__ rowsum_part, const float* __restrict__ diag,
                         const float* __restrict__ pos_sum, const float* __restrict__ pos_cnt,
                         float* out, int N) {
    const int i    = blockIdx.x * blockDim.x + threadIdx.x;
    const int lane = threadIdx.x & 31;
    float li = 0.0f;
    if (i < N && pos_cnt[i] > 0.0f) {
        float rs = 0.0f;
#pragma unroll
        for (int p = 0; p < JCHUNKS; ++p) rs += rowsum_part[(size_t)p * N + i];
        const float dg       = diag[i];
        const float corr_pos = dg + pos_sum[i];
        const float corr_neg = rs - dg;
        li = -__logf(corr_pos / (corr_pos + corr_neg));
    }
    li += __shfl_xor(li, 16);
    li += __shfl_xor(li, 8);
    li += __shfl_xor(li, 4);
    li += __shfl_xor(li, 2);
    li += __shfl_xor(li, 1);
    if (lane == 0) atomicAdd(out, li / (float)N);
}

extern "C" void kernel_launch(void* const* d_in, const int* in_sizes, int n_in,
                              void* d_out, int out_size, void* d_ws, size_t ws_size,
                              hipStream_t stream) {
    const float* Z     = (const float*)d_in[0];
    const int*   ppair = (const int*)d_in[1];   // positive_pairs; negative_pairs (d_in[2]) unused
    const int N = in_sizes[0] / D_DIM;          // 8192
    const int P = in_sizes[1] / 2;              // 81920

    float* Zn          = (float*)d_ws;                      // N*128 floats
    float* rowsum_part = Zn + (size_t)N * D_DIM;            // JCHUNKS*N floats
    float* diag        = rowsum_part + (size_t)JCHUNKS * N; // N
    float* pos_sum     = diag + N;                          // N
    float* pos_cnt     = pos_sum + N;                       // N
    float* out         = (float*)d_out;

    lts_init_kernel<<<(N + 255) / 256, 256, 0, stream>>>(pos_sum, pos_cnt, out, N);
    lts_normalize_kernel<<<(N + 7) / 8, 256, 0, stream>>>(Z, Zn, diag, N);
    lts_pairs_kernel<<<(P + 7) / 8, 256, 0, stream>>>(Zn, ppair, pos_sum, pos_cnt, N, P);
    dim3 ggrid(N / 128, JCHUNKS);
    lts_gram_rowsum_kernel<<<ggrid, 256, 0, stream>>>(Zn, rowsum_part, N);
    lts_finalize_kernel<<<(N + 255) / 256, 256, 0, stream>>>(rowsum_part, diag, pos_sum, pos_cnt, out, N);
}